// BiDAFAttention_8031588844178
// MI455X (gfx1250) — compile-verified
//
#include <hip/hip_runtime.h>
#include <cstdint>

// ---------------------------------------------------------------------------
// BiDAF attention for MI455X (gfx1250, wave32, WMMA).
// Shapes: B=64, C=512, Q=64, D=256, fp32 everywhere.
//
// Roofline: ~2.2 GFLOP of matmul vs ~165 MB of traffic (128 MB of it the
// output tensor) -> memory bound, floor ~7us at 23.3 TB/s. So: keep full fp32
// precision using V_WMMA_F32_16X16X4_F32 (far below its ceiling), stage
// inputs once in LDS via gfx1250 async global->LDS, keep sim/softmax/
// attention entirely on-chip, and write the 128 MB output exactly once with
// coalesced stores.
// ---------------------------------------------------------------------------

#define B_  64
#define C_  512
#define Q_  64
#define D_  256
#define CT  64              // context rows per workgroup tile
#define LDC (D_ + 4)        // padded LDS row stride -> conflict-free col gathers
#define LDS_S (Q_ + 4)      // padded stride for the 64x64 sim/attention tile

#define NEG_F (-3.4028234663852886e38f)   // jnp.finfo(f32).min

typedef float v2f __attribute__((ext_vector_type(2)));
typedef float v8f __attribute__((ext_vector_type(8)));

#if defined(__gfx1250__) && __has_builtin(__builtin_amdgcn_global_load_async_to_lds_b128)
#define HAVE_ASYNC_LDS 1
// Builtin signature (from hipcc diagnostic): param0 is
//   int __attribute__((vector_size(16))) __device__*   (AS1 = global)
typedef int v4i __attribute__((vector_size(16)));
typedef __attribute__((address_space(1))) v4i g_v4i;
typedef __attribute__((address_space(3))) v4i l_v4i;
#else
#define HAVE_ASYNC_LDS 0
#endif

__device__ __forceinline__ void wait_asynccnt0() {
#if defined(__gfx1250__)
#if __has_builtin(__builtin_amdgcn_s_wait_asynccnt)
    __builtin_amdgcn_s_wait_asynccnt(0);
#else
    asm volatile("s_wait_asynccnt 0x0" ::: "memory");
#endif
#endif
}

// ---------------------------------------------------------------------------
// Kernel A: per (batch, 64-row context tile)
//   sim = rowc[c] + rowq[q] + (ctx*w_m) @ q^T      (WMMA f32 16x16x4)
//   a   = softmax_q(mask_q(sim)); raw max_q(sim) -> ws
//   c2q = a @ q                                    (WMMA f32 16x16x4)
//   out sections 0..2: ctx, c2q, ctx*c2q
// ---------------------------------------------------------------------------
__global__ __launch_bounds__(256)
void bidaf_sim_c2q(const float* __restrict__ ctx, const float* __restrict__ qst,
                   const int* __restrict__ qmask_g, const float* __restrict__ w,
                   float* __restrict__ out, float* __restrict__ simmax_ws)
{
    __shared__ float Cs[CT * LDC];      // context tile          (65 KB)
    __shared__ float Qs[Q_ * LDC];      // question (this batch) (65 KB)
    __shared__ float S [CT * LDS_S];    // sim / attention tile  (17 KB)
    __shared__ float Wc[D_], Wq[D_], Wm[D_];
    __shared__ float rowc[CT], rowq[Q_];
    __shared__ float red[256];
    __shared__ int   qm[Q_];

    const int tid = threadIdx.x;
    const int b   = blockIdx.x;
    const int c0  = blockIdx.y * CT;

    // ---- stage weights / masks ----
    for (int i = tid; i < D_; i += 256) {
        Wc[i] = w[i];
        Wq[i] = w[D_ + i];
        Wm[i] = w[2 * D_ + i];
    }
    if (tid < Q_) qm[tid] = qmask_g[b * Q_ + tid];

    // ---- stage ctx tile + question into LDS ----
    const float4* ctx4 = (const float4*)(ctx + ((size_t)b * C_ + c0) * D_);
    const float4* qst4 = (const float4*)(qst + (size_t)b * Q_ * D_);
#if HAVE_ASYNC_LDS
    // gfx1250 direct async global->LDS (ASYNCcnt): no VGPR round-trip.
    for (int i = tid; i < CT * (D_ / 4); i += 256) {
        const int r = i >> 6, c4 = i & 63;          // 64 float4 per row
        __builtin_amdgcn_global_load_async_to_lds_b128(
            (g_v4i*)(ctx4 + i), (l_v4i*)&Cs[r * LDC + c4 * 4], 0, 0);
        __builtin_amdgcn_global_load_async_to_lds_b128(
            (g_v4i*)(qst4 + i), (l_v4i*)&Qs[r * LDC + c4 * 4], 0, 0);
    }
    wait_asynccnt0();
#else
    for (int i = tid; i < CT * (D_ / 4); i += 256) {
        const int r = i >> 6, c4 = i & 63;          // 64 float4 per row
        *(float4*)&Cs[r * LDC + c4 * 4] = ctx4[i];
        *(float4*)&Qs[r * LDC + c4 * 4] = qst4[i];
    }
#endif
    __syncthreads();

    // ---- rank-1 terms: rowc = Cs.Wc, rowq = Qs.Wq (4 partials per row) ----
    {
        const int r = tid & 63, p = tid >> 6;
        float sc = 0.f, sq = 0.f;
        for (int d = p * 64; d < p * 64 + 64; ++d) {
            sc += Cs[r * LDC + d] * Wc[d];
            sq += Qs[r * LDC + d] * Wq[d];
        }
        red[tid] = sc;
        __syncthreads();
        if (tid < 64) rowc[tid] = red[tid] + red[tid + 64] + red[tid + 128] + red[tid + 192];
        __syncthreads();
        red[tid] = sq;
        __syncthreads();
        if (tid < 64) rowq[tid] = red[tid] + red[tid + 64] + red[tid + 128] + red[tid + 192];
        __syncthreads();
    }

    const int wid  = tid >> 5;      // wave id 0..7
    const int lane = tid & 31;
    const int lr   = lane & 15;     // row/col within half-wave
    const int lh   = lane >> 4;     // half-wave select

    // fp32 A/B fragment layout (ISA 7.12.2): VGPR v holds K=v for lanes 0-15,
    // K=v+2 for lanes 16-31; A lane index = M row, B lane index = N col.
    // C/D: VGPR i -> rows i (lanes 0-15) and i+8 (lanes 16-31), col = lane&15.

    // ---- GEMM1: sim(64x64) = (Cs*Wm) @ Qs^T, K=D=256 ----
    for (int t = wid; t < 16; t += 8) {             // 4x4 grid of 16x16 tiles
        const int tm = (t >> 2) * 16;
        const int tn = (t & 3) * 16;
        v8f acc = {};
        for (int k = 0; k < D_; k += 4) {
            const int kk = k + 2 * lh;
            v2f a, bb;
            a.x  = Cs[(tm + lr) * LDC + kk]     * Wm[kk];
            a.y  = Cs[(tm + lr) * LDC + kk + 1] * Wm[kk + 1];
            // B[k=d][n=q] = question[q][d]:
            bb.x = Qs[(tn + lr) * LDC + kk];
            bb.y = Qs[(tn + lr) * LDC + kk + 1];
            acc = __builtin_amdgcn_wmma_f32_16x16x4_f32(
                      false, a, false, bb, (short)0, acc, false, false);
        }
        const float cq = rowq[tn + lr];
        #pragma unroll
        for (int i = 0; i < 8; ++i) {
            const int r = tm + i + lh * 8;
            S[r * LDS_S + tn + lr] = acc[i] + rowc[r] + cq;
        }
    }
    __syncthreads();

    // ---- softmax over q (question-masked) + raw row max ----
    if (tid < CT) {
        const int r = tid;
        float rawmax = NEG_F, mmax = NEG_F;
        for (int q = 0; q < Q_; ++q) {
            const float v = S[r * LDS_S + q];
            rawmax = fmaxf(rawmax, v);
            if (qm[q]) mmax = fmaxf(mmax, v);
        }
        simmax_ws[(size_t)b * C_ + c0 + r] = rawmax;
        float sum = 0.f;
        for (int q = 0; q < Q_; ++q) {
            const float e = qm[q] ? __expf(S[r * LDS_S + q] - mmax) : 0.f;
            S[r * LDS_S + q] = e;
            sum += e;
        }
        const float inv = 1.f / sum;
        for (int q = 0; q < Q_; ++q) S[r * LDS_S + q] *= inv;
    }
    __syncthreads();

    // ---- GEMM2: c2q(64x256) = a(64x64) @ q(64x256), K=Q=64; fused output ----
    for (int j = 0; j < 8; ++j) {
        const int t  = wid * 8 + j;                 // 64 tiles: 4 row x 16 col
        const int tm = (t >> 4) * 16;
        const int tn = (t & 15) * 16;
        v8f acc = {};
        for (int k = 0; k < Q_; k += 4) {
            const int kk = k + 2 * lh;
            v2f a, bb;
            a.x  = S[(tm + lr) * LDS_S + kk];
            a.y  = S[(tm + lr) * LDS_S + kk + 1];
            // B[k=q][n=d] = question[q][d]:
            bb.x = Qs[(kk)     * LDC + tn + lr];
            bb.y = Qs[(kk + 1) * LDC + tn + lr];
            acc = __builtin_amdgcn_wmma_f32_16x16x4_f32(
                      false, a, false, bb, (short)0, acc, false, false);
        }
        #pragma unroll
        for (int i = 0; i < 8; ++i) {
            const int r = tm + i + lh * 8;
            const int d = tn + lr;
            const float cv  = Cs[r * LDC + d];
            const float c2q = acc[i];
            float* o = out + ((size_t)b * C_ + c0 + r) * (4 * D_);
            o[d]          = cv;
            o[D_ + d]     = c2q;
            o[2 * D_ + d] = cv * c2q;
        }
    }
}

// ---------------------------------------------------------------------------
// Kernel B: per batch — softmax_c(mask_c(sim_max)) then q2c[d] = sum_c b_c*ctx
// ---------------------------------------------------------------------------
__global__ __launch_bounds__(256)
void bidaf_q2c(const float* __restrict__ ctx, const int* __restrict__ cmask,
               const float* __restrict__ simmax_ws, float* __restrict__ q2c_ws)
{
    __shared__ float wts[C_];
    __shared__ float red[256];
    const int b = blockIdx.x, tid = threadIdx.x;

    const float v0 = cmask[b * C_ + tid]       ? simmax_ws[(size_t)b * C_ + tid]       : NEG_F;
    const float v1 = cmask[b * C_ + tid + 256] ? simmax_ws[(size_t)b * C_ + tid + 256] : NEG_F;

    red[tid] = fmaxf(v0, v1);
    __syncthreads();
    for (int s = 128; s > 0; s >>= 1) {
        if (tid < s) red[tid] = fmaxf(red[tid], red[tid + s]);
        __syncthreads();
    }
    const float M = red[0];
    __syncthreads();

    const float e0 = __expf(v0 - M), e1 = __expf(v1 - M);
    wts[tid] = e0; wts[tid + 256] = e1;
    red[tid] = e0 + e1;
    __syncthreads();
    for (int s = 128; s > 0; s >>= 1) {
        if (tid < s) red[tid] += red[tid + s];
        __syncthreads();
    }
    const float inv = 1.f / red[0];
    __syncthreads();
    wts[tid] *= inv; wts[tid + 256] *= inv;
    __syncthreads();

    // tid == d; each iteration reads one contiguous 1KB ctx row (coalesced)
    float acc = 0.f;
    const float* cb = ctx + (size_t)b * C_ * D_;
    for (int c = 0; c < C_; ++c) acc += wts[c] * cb[c * D_ + tid];
    q2c_ws[b * D_ + tid] = acc;
}

// ---------------------------------------------------------------------------
// Kernel C: out section 3 = ctx * broadcast(q2c), float4 streams
// ---------------------------------------------------------------------------
__global__ __launch_bounds__(256)
void bidaf_gate_q2c(const float* __restrict__ ctx, const float* __restrict__ q2c_ws,
                    float* __restrict__ out)
{
    const size_t i = (size_t)blockIdx.x * 256 + threadIdx.x;  // float4 idx over B*C*D/4
    const int d4 = (int)(i & 63);
    const int c  = (int)((i >> 6) & 511);
    const int b  = (int)(i >> 15);                            // C*D/4 = 2^15
    const float4 cv = ((const float4*)ctx)[i];
    const float4 qv = *(const float4*)&q2c_ws[b * D_ + d4 * 4];
    float4 r;
    r.x = cv.x * qv.x; r.y = cv.y * qv.y; r.z = cv.z * qv.z; r.w = cv.w * qv.w;
    *(float4*)&out[((size_t)b * C_ + c) * (4 * D_) + 3 * D_ + d4 * 4] = r;
}

// ---------------------------------------------------------------------------
extern "C" void kernel_launch(void* const* d_in, const int* in_sizes, int n_in,
                              void* d_out, int out_size, void* d_ws, size_t ws_size,
                              hipStream_t stream)
{
    (void)in_sizes; (void)n_in; (void)out_size; (void)ws_size;
    const float* ctx   = (const float*)d_in[0];
    const float* qst   = (const float*)d_in[1];
    const int*   cmask = (const int*)d_in[2];
    const int*   qmask = (const int*)d_in[3];
    const float* w     = (const float*)d_in[4];
    float* out = (float*)d_out;

    float* simmax = (float*)d_ws;             // B*C floats
    float* q2c    = simmax + B_ * C_;         // B*D floats

    dim3 gA(B_, C_ / CT);
    bidaf_sim_c2q<<<gA, 256, 0, stream>>>(ctx, qst, qmask, w, out, simmax);
    bidaf_q2c<<<B_, 256, 0, stream>>>(ctx, cmask, simmax, q2c);
    bidaf_gate_q2c<<<(B_ * C_ * D_ / 4) / 256, 256, 0, stream>>>(ctx, q2c, out);
}